// GraphDiffNetBipartite_46720654246029
// MI455X (gfx1250) — compile-verified
//
#include <hip/hip_runtime.h>
#include <hip/hip_bf16.h>
#include <math.h>

// ---------------------------------------------------------------------------
// MI455X / gfx1250 implementation. Dense GEMMs use V_WMMA_F32_16X16X4_F32
// (full f32 precision, wave32). Each wave computes a 16x64 output stripe:
// one v2f A-fragment load feeds 4 WMMAs; B (the KxN=64 weight matrix) is
// staged transposed into LDS once per workgroup so B fragments are single
// ds_load_b64 reads. Irregular GCN parts use VALU + global f32 atomics.
// ---------------------------------------------------------------------------

typedef __attribute__((ext_vector_type(2))) float v2f;
typedef __attribute__((ext_vector_type(8))) float v8f;

__device__ __forceinline__ float dinvf(float d) {
    return d > 0.0f ? __frsqrt_rn(d) : 0.0f;
}

#define WMMA4(aval, kkv)                                                          \
    {                                                                             \
        v2f b0 = *(const v2f*)(&LWt[(lm +  0) * K + (kkv)]);                      \
        v2f b1 = *(const v2f*)(&LWt[(lm + 16) * K + (kkv)]);                      \
        v2f b2 = *(const v2f*)(&LWt[(lm + 32) * K + (kkv)]);                      \
        v2f b3 = *(const v2f*)(&LWt[(lm + 48) * K + (kkv)]);                      \
        acc0 = __builtin_amdgcn_wmma_f32_16x16x4_f32(false, aval, false, b0, (short)0, acc0, false, false); \
        acc1 = __builtin_amdgcn_wmma_f32_16x16x4_f32(false, aval, false, b1, (short)0, acc1, false, false); \
        acc2 = __builtin_amdgcn_wmma_f32_16x16x4_f32(false, aval, false, b2, (short)0, acc2, false, false); \
        acc3 = __builtin_amdgcn_wmma_f32_16x16x4_f32(false, aval, false, b3, (short)0, acc3, false, false); \
    }

// ---------------------------------------------------------------------------
// Generic WMMA GEMM: out(M x 64) = [A0 | A1](M x K) @ W(K x 64) (+bias)(relu)
// A split at column `ksplit` (multiple of 4) to fuse concat([diff_x, x_diff]).
// blockDim = 128 (4 waves); wave w of block b computes rows
// [(4b+w)*16, +16) x all 64 columns.  Requires 16 | M, 4 | K, K <= 128.
// ---------------------------------------------------------------------------
__global__ __launch_bounds__(128) void gemm64_wmma(
    const float* __restrict__ A0, int lda0,
    const float* __restrict__ A1, int lda1, int ksplit,
    const float* __restrict__ W,
    const float* __restrict__ bias,
    float* __restrict__ out,
    int M, int K, int do_relu, int accumulate)
{
    __shared__ __align__(16) float LWt[64 * 128];   // W transposed: LWt[n*K + k]
    const int tid = threadIdx.x;
    for (int idx = tid; idx < K * 64; idx += 128) {
        int k = idx >> 6, n = idx & 63;
        LWt[n * K + k] = W[idx];
    }
    __syncthreads();

    const int wave = tid >> 5;
    const int lane = tid & 31;
    const int lm = lane & 15;
    const int hi = lane >> 4;              // lane half selects K pair
    const int mtile = blockIdx.x * 4 + wave;
    if (mtile * 16 >= M) return;
    const int arow = mtile * 16 + lm;
    const float* a0p = A0 + (size_t)arow * lda0;
    const float* a1p = A1 + (size_t)arow * lda1 - ksplit;
    const int koff = hi << 1;

    v8f acc0 = {}, acc1 = {}, acc2 = {}, acc3 = {};
    int k = 0;
    const int kend0 = (ksplit < K) ? ksplit : K;
    for (; k < kend0; k += 4) {
        const int kk = k + koff;
        v2f a = *(const v2f*)(a0p + kk);
        WMMA4(a, kk);
    }
    for (; k < K; k += 4) {
        const int kk = k + koff;
        v2f a = *(const v2f*)(a1p + kk);
        WMMA4(a, kk);
    }

#pragma unroll
    for (int t = 0; t < 4; ++t) {
        const int n = t * 16 + lm;
        const float bv = bias ? bias[n] : 0.0f;
        const v8f acc = (t == 0) ? acc0 : (t == 1) ? acc1 : (t == 2) ? acc2 : acc3;
#pragma unroll
        for (int r = 0; r < 8; ++r) {
            const int row = mtile * 16 + r + (hi << 3);
            float v = acc[r] + bv;
            if (do_relu) v = fmaxf(v, 0.0f);
            const size_t o = (size_t)row * 64 + n;
            if (accumulate) out[o] += v; else out[o] = v;
        }
    }
}

// ---------------------------------------------------------------------------
// Split-K WMMA GEMM for x_spec = evecs^T @ (mass * diff_x):
//   M = 128 (eig), N = 64, K = N_SURF (30000); partials via f32 atomics.
// ET is evecs pre-transposed (128 x NS) so A fragments are contiguous v2f.
// grid.x = 2 * num_chunks (2 blocks x 4 waves cover the 8 row-tiles).
// ---------------------------------------------------------------------------
__global__ __launch_bounds__(128) void spec_gemm1_wmma(
    const float* __restrict__ ET,     // (128, NS)
    const float* __restrict__ mx,     // (NS, 64)
    float* __restrict__ xspec,        // (128, 64)
    int NS, int kchunk)
{
    const int wave = threadIdx.x >> 5;
    const int lane = threadIdx.x & 31;
    const int lm = lane & 15, hi = lane >> 4;
    const int mhalf = blockIdx.x & 1;
    const int chunk = blockIdx.x >> 1;
    const int mtile = mhalf * 4 + wave;        // 0..7 over 128 eig rows
    const int m = mtile * 16 + lm;
    const int k0 = chunk * kchunk;
    int k1 = k0 + kchunk; if (k1 > NS) k1 = NS;
    const float* ap = ET + (size_t)m * NS;
    const int koff = hi << 1;

    v8f acc0 = {}, acc1 = {}, acc2 = {}, acc3 = {};
    for (int k = k0; k < k1; k += 4) {
        const int kk = k + koff;
        v2f a = *(const v2f*)(ap + kk);
        const float* m0 = mx + (size_t)kk * 64;
        v2f b0, b1, b2, b3;
        b0.x = m0[lm +  0]; b0.y = m0[64 + lm +  0];
        b1.x = m0[lm + 16]; b1.y = m0[64 + lm + 16];
        b2.x = m0[lm + 32]; b2.y = m0[64 + lm + 32];
        b3.x = m0[lm + 48]; b3.y = m0[64 + lm + 48];
        acc0 = __builtin_amdgcn_wmma_f32_16x16x4_f32(false, a, false, b0, (short)0, acc0, false, false);
        acc1 = __builtin_amdgcn_wmma_f32_16x16x4_f32(false, a, false, b1, (short)0, acc1, false, false);
        acc2 = __builtin_amdgcn_wmma_f32_16x16x4_f32(false, a, false, b2, (short)0, acc2, false, false);
        acc3 = __builtin_amdgcn_wmma_f32_16x16x4_f32(false, a, false, b3, (short)0, acc3, false, false);
    }
#pragma unroll
    for (int t = 0; t < 4; ++t) {
        const int n = t * 16 + lm;
        const v8f acc = (t == 0) ? acc0 : (t == 1) ? acc1 : (t == 2) ? acc2 : acc3;
#pragma unroll
        for (int r = 0; r < 8; ++r) {
            const int row = mtile * 16 + r + (hi << 3);
            atomicAdd(&xspec[(size_t)row * 64 + n], acc[t == 0 ? r : r]);
            (void)acc;
        }
    }
}

// ---------------------------------------------------------------------------
// Small elementwise / irregular kernels
// ---------------------------------------------------------------------------
__global__ void fill_f32(float* p, int n, float v) {
    int t = blockIdx.x * blockDim.x + threadIdx.x;
    if (t < n) p[t] = v;
}

__global__ void transpose_evecs(float* __restrict__ ET, const float* __restrict__ evecs,
                                int NS)
{
    int t = blockIdx.x * blockDim.x + threadIdx.x;
    if (t >= NS * 128) return;
    int n = t >> 7, m = t & 127;
    ET[(size_t)m * NS + n] = evecs[t];
}

__global__ void lin_small(const float* __restrict__ A, const float* __restrict__ W,
                          const float* __restrict__ b, float* __restrict__ out,
                          int M, int K)
{
    int t = blockIdx.x * blockDim.x + threadIdx.x;
    if (t >= M * 64) return;
    int m = t >> 6, n = t & 63;
    float s = b[n];
    for (int k = 0; k < K; ++k) s += A[(size_t)m * K + k] * W[(size_t)k * 64 + n];
    out[t] = s;
}

__global__ void scale_mass(float* __restrict__ out, const float* __restrict__ x,
                           const float* __restrict__ mass, int n)
{
    int t = blockIdx.x * blockDim.x + threadIdx.x;
    if (t < n) out[t] = mass[t >> 6] * x[t];
}

__global__ void bw_kernel(float* __restrict__ bw, const float* __restrict__ xspec,
                          const float* __restrict__ evals, const float* __restrict__ tvec)
{
    int t = blockIdx.x * blockDim.x + threadIdx.x;
    if (t >= 128 * 64) return;
    int k = t >> 6, n = t & 63;
    float tt = fmaxf(tvec[n], 1e-8f);
    bw[t] = __expf(-evals[k] * tt) * xspec[t];
}

__global__ void deg_accum(float* __restrict__ deg, const int* __restrict__ dst,
                          const float* __restrict__ w, int E)
{
    int e = blockIdx.x * blockDim.x + threadIdx.x;
    if (e < E) atomicAdd(&deg[dst[e]], w ? w[e] : 1.0f);
}

__global__ void gcn_init(float* __restrict__ out, const float* __restrict__ h,
                         const float* __restrict__ deg, const float* __restrict__ b,
                         int lo, int hi)
{
    int t = blockIdx.x * blockDim.x + threadIdx.x;
    int nrows = hi - lo;
    if (t >= nrows * 64) return;
    int v = lo + (t >> 6), c = t & 63;
    float dv = dinvf(deg[v]);
    out[(size_t)v * 64 + c] = dv * dv * h[(size_t)v * 64 + c] + b[c];
}

__global__ void gcn_scatter(float* __restrict__ out, const float* __restrict__ h,
                            const float* __restrict__ deg,
                            const int* __restrict__ src, const int* __restrict__ dst,
                            const float* __restrict__ w, int E)
{
    int t = blockIdx.x * blockDim.x + threadIdx.x;
    if (t >= E * 16) return;
    int e = t >> 4;
    int c = (t & 15) << 2;
    int s = src[e], d = dst[e];
    float nm = dinvf(deg[s]) * dinvf(deg[d]) * (w ? w[e] : 1.0f);
    const float* hs = h + (size_t)s * 64 + c;
    float* od = out + (size_t)d * 64 + c;
    atomicAdd(od + 0, nm * hs[0]);
    atomicAdd(od + 1, nm * hs[1]);
    atomicAdd(od + 2, nm * hs[2]);
    atomicAdd(od + 3, nm * hs[3]);
}

__global__ void relu_k(float* x, int n) {
    int t = blockIdx.x * blockDim.x + threadIdx.x;
    if (t < n) x[t] = fmaxf(x[t], 0.0f);
}

__global__ void matvec64(float* __restrict__ out, const float* __restrict__ x,
                         const float* __restrict__ w, int M)
{
    int v = blockIdx.x * blockDim.x + threadIdx.x;
    if (v >= M) return;
    float s = 0.0f;
    for (int c = 0; c < 64; ++c) s += x[(size_t)v * 64 + c] * w[c];
    out[v] = s;
}

__global__ void final_init(float* __restrict__ out,
                           const float* __restrict__ hv1, const float* __restrict__ hv2,
                           const float* __restrict__ degA, const float* __restrict__ degB,
                           const float* __restrict__ bA, const float* __restrict__ bB,
                           int NS, int NT)
{
    int v = blockIdx.x * blockDim.x + threadIdx.x;
    if (v >= NT) return;
    if (v < NS) {
        float dv = dinvf(degA[v]);
        out[v] = dv * dv * hv1[v] + bA[0];
    } else {
        float dv = dinvf(degB[v]);
        out[v] = dv * dv * hv2[v] + bB[0];
    }
}

__global__ void final_scatter(float* __restrict__ out, const float* __restrict__ hv,
                              const float* __restrict__ deg,
                              const int* __restrict__ src, const int* __restrict__ dst,
                              const float* __restrict__ w, int E)
{
    int e = blockIdx.x * blockDim.x + threadIdx.x;
    if (e >= E) return;
    int s = src[e], d = dst[e];
    float nm = dinvf(deg[s]) * dinvf(deg[d]) * w[e];
    atomicAdd(&out[d], nm * hv[s]);
}

// ---------------------------------------------------------------------------
// Host orchestration
// ---------------------------------------------------------------------------
static inline dim3 gblk(long n, int bs) { return dim3((unsigned)((n + bs - 1) / bs)); }

extern "C" void kernel_launch(void* const* d_in, const int* in_sizes, int n_in,
                              void* d_out, int out_size, void* d_ws, size_t ws_size,
                              hipStream_t stream)
{
    const float* x_surf    = (const float*)d_in[0];
    const float* x_graph   = (const float*)d_in[1];
    const float* mass      = (const float*)d_in[2];
    const float* evals     = (const float*)d_in[3];
    const float* evecs     = (const float*)d_in[4];
    const float* bi_w      = (const float*)d_in[5];
    const float* lin1_w    = (const float*)d_in[6];
    const float* lin1_b    = (const float*)d_in[7];
    const float* lin2_w    = (const float*)d_in[8];
    const float* lin2_b    = (const float*)d_in[9];
    const float* diff_time = (const float*)d_in[10];
    const float* mlp_w0    = (const float*)d_in[11];
    const float* mlp_b0    = (const float*)d_in[12];
    const float* mlp_w1    = (const float*)d_in[13];
    const float* mlp_b1    = (const float*)d_in[14];
    const float* mlp_w2    = (const float*)d_in[15];
    const float* mlp_b2    = (const float*)d_in[16];
    const float* gcn_w1    = (const float*)d_in[17];
    const float* gcn_b1    = (const float*)d_in[18];
    const float* gcn_w2    = (const float*)d_in[19];
    const float* gcn_b2    = (const float*)d_in[20];
    const float* gs_w      = (const float*)d_in[21];
    const float* gs_b      = (const float*)d_in[22];
    const float* gs_w_last = (const float*)d_in[23];
    const float* gs_b_last = (const float*)d_in[24];
    const float* sg_w      = (const float*)d_in[25];
    const float* sg_b      = (const float*)d_in[26];
    const float* sg_w_last = (const float*)d_in[27];
    const float* sg_b_last = (const float*)d_in[28];
    const int*   gedge     = (const int*)d_in[29];
    const int*   bi_surf   = (const int*)d_in[30];
    const int*   bi_graph  = (const int*)d_in[31];

    const int NS  = in_sizes[2];       // 30000
    const int KE  = in_sizes[3];       // 128
    const int NG  = in_sizes[1] / 20;  // 12000
    const int NT  = NS + NG;           // 42000
    const int EG  = in_sizes[29] / 2;  // 192000
    const int EBI = in_sizes[5];
    float* outp = (float*)d_out;

    // ---- workspace carve (floats) -----------------------------------------
    float* ws = (float*)d_ws;
    size_t off = 0;
    float* XA    = ws + off; off += (size_t)NT * 64;   // x_cat ping
    float* XB    = ws + off; off += (size_t)NT * 64;   // x_cat pong
    float* XDIFF = ws + off; off += (size_t)NS * 64;   // x_diffuse
    float* H1    = ws + off; off += (size_t)NT * 64;   // scratch / mlp h / hv1
    float* H2    = ws + off; off += (size_t)NT * 64;   // scratch / mass*x / hv2
    float* GT    = ws + off; off += (size_t)NG * 64;   // graph gemm out
    float* HG    = ws + off; off += (size_t)NG * 64;   // graph hidden
    float* ET    = ws + off; off += (size_t)KE * NS;   // evecs^T (128 x NS)
    float* XSPEC = ws + off; off += (size_t)KE * 64;
    float* BW    = ws + off; off += (size_t)KE * 64;
    float* DEGG  = ws + off; off += (size_t)NG;
    float* DEGA  = ws + off; off += (size_t)NT;
    float* DEGB  = ws + off; off += (size_t)NT;
    (void)ws_size;

    const int BS = 256;

    // ---- one-time prep ----------------------------------------------------
    transpose_evecs<<<gblk((long)NS * 128, BS), BS, 0, stream>>>(ET, evecs, NS);

    fill_f32<<<gblk(NG, BS), BS, 0, stream>>>(DEGG, NG, 1.0f);
    deg_accum<<<gblk(EG, BS), BS, 0, stream>>>(DEGG, gedge + EG, nullptr, EG);
    fill_f32<<<gblk(NT, BS), BS, 0, stream>>>(DEGA, NT, 1.0f);
    deg_accum<<<gblk(EBI, BS), BS, 0, stream>>>(DEGA, bi_surf, bi_w, EBI);
    fill_f32<<<gblk(NT, BS), BS, 0, stream>>>(DEGB, NT, 1.0f);
    deg_accum<<<gblk(EBI, BS), BS, 0, stream>>>(DEGB, bi_graph, bi_w, EBI);

    lin_small<<<gblk((long)NS * 64, BS), BS, 0, stream>>>(x_surf, lin1_w, lin1_b, XA, NS, 5);
    lin_small<<<gblk((long)NG * 64, BS), BS, 0, stream>>>(x_graph, lin2_w, lin2_b,
                                                          XA + (size_t)NS * 64, NG, 20);

    const int KCH = 600;
    const int NCH = (NS + KCH - 1) / KCH;
    const int GB_NS = (NS / 16 + 3) / 4;   // gemm64 grid for M=NS
    const int GB_NG = (NG / 16 + 3) / 4;
    const int GB_NT = (NT / 16 + 3) / 4;

    for (int i = 0; i < 4; ++i) {
        float* cur = (i & 1) ? XB : XA;
        float* nxt = (i & 1) ? XA : XB;
        float* surf = cur;
        float* gcur = cur + (size_t)NS * 64;

        // ---- spectral diffusion ------------------------------------------
        scale_mass<<<gblk((long)NS * 64, BS), BS, 0, stream>>>(H2, surf, mass, NS * 64);
        fill_f32<<<gblk(KE * 64, BS), BS, 0, stream>>>(XSPEC, KE * 64, 0.0f);
        spec_gemm1_wmma<<<2 * NCH, 128, 0, stream>>>(ET, H2, XSPEC, NS, KCH);
        bw_kernel<<<gblk(KE * 64, BS), BS, 0, stream>>>(BW, XSPEC, evals, diff_time + i * 64);
        // x_diffuse = evecs @ BW
        gemm64_wmma<<<GB_NS, 128, 0, stream>>>(evecs, KE, evecs, KE, KE,
                                               BW, nullptr, XDIFF, NS, KE, 0, 0);

        // ---- MLP on surface (feat = [diff_x | x_diffuse]) -----------------
        gemm64_wmma<<<GB_NS, 128, 0, stream>>>(surf, 64, XDIFF, 64, 64,
                                               mlp_w0 + (size_t)i * 128 * 64,
                                               mlp_b0 + i * 64, H1, NS, 128, 1, 0);
        gemm64_wmma<<<GB_NS, 128, 0, stream>>>(H1, 64, H1, 64, 64,
                                               mlp_w1 + (size_t)i * 64 * 64,
                                               mlp_b1 + i * 64, H2, NS, 64, 1, 0);
        gemm64_wmma<<<GB_NS, 128, 0, stream>>>(H2, 64, H2, 64, 64,
                                               mlp_w2 + (size_t)i * 64 * 64,
                                               mlp_b2 + i * 64, surf, NS, 64, 0, 1);

        // ---- graph GCN (2 layers) ----------------------------------------
        gemm64_wmma<<<GB_NG, 128, 0, stream>>>(gcur, 64, gcur, 64, 64,
                                               gcn_w1 + (size_t)i * 64 * 64,
                                               nullptr, GT, NG, 64, 0, 0);
        gcn_init<<<gblk((long)NG * 64, BS), BS, 0, stream>>>(HG, GT, DEGG, gcn_b1 + i * 64, 0, NG);
        gcn_scatter<<<gblk((long)EG * 16, BS), BS, 0, stream>>>(HG, GT, DEGG,
                                                                gedge, gedge + EG, nullptr, EG);
        relu_k<<<gblk((long)NG * 64, BS), BS, 0, stream>>>(HG, NG * 64);
        gemm64_wmma<<<GB_NG, 128, 0, stream>>>(HG, 64, HG, 64, 64,
                                               gcn_w2 + (size_t)i * 64 * 64,
                                               nullptr, GT, NG, 64, 0, 0);
        gcn_init<<<gblk((long)NG * 64, BS), BS, 0, stream>>>(gcur, GT, DEGG, gcn_b2 + i * 64, 0, NG);
        gcn_scatter<<<gblk((long)EG * 16, BS), BS, 0, stream>>>(gcur, GT, DEGG,
                                                                gedge, gedge + EG, nullptr, EG);

        // ---- bipartite cross convs ---------------------------------------
        if (i < 3) {
            gemm64_wmma<<<GB_NT, 128, 0, stream>>>(cur, 64, cur, 64, 64,
                                                   gs_w + (size_t)i * 64 * 64,
                                                   nullptr, H1, NT, 64, 0, 0);
            gemm64_wmma<<<GB_NT, 128, 0, stream>>>(cur, 64, cur, 64, 64,
                                                   sg_w + (size_t)i * 64 * 64,
                                                   nullptr, H2, NT, 64, 0, 0);
            // new_surf (rows [0,NS)) from graph->surf edges
            gcn_init<<<gblk((long)NS * 64, BS), BS, 0, stream>>>(nxt, H1, DEGA, gs_b + i * 64, 0, NS);
            gcn_scatter<<<gblk((long)EBI * 16, BS), BS, 0, stream>>>(nxt, H1, DEGA,
                                                                     bi_graph, bi_surf, bi_w, EBI);
            // new_graph (rows [NS,NT)) from surf->graph edges
            gcn_init<<<gblk((long)NG * 64, BS), BS, 0, stream>>>(nxt, H2, DEGB, sg_b + i * 64, NS, NT);
            gcn_scatter<<<gblk((long)EBI * 16, BS), BS, 0, stream>>>(nxt, H2, DEGB,
                                                                     bi_surf, bi_graph, bi_w, EBI);
        } else {
            // last block: C_OUT = 1 -> matvec + scalar scatter into d_out
            matvec64<<<gblk(NT, BS), BS, 0, stream>>>(H1, cur, gs_w_last, NT);
            matvec64<<<gblk(NT, BS), BS, 0, stream>>>(H2, cur, sg_w_last, NT);
            final_init<<<gblk(NT, BS), BS, 0, stream>>>(outp, H1, H2, DEGA, DEGB,
                                                        gs_b_last, sg_b_last, NS, NT);
            final_scatter<<<gblk(EBI, BS), BS, 0, stream>>>(outp, H1, DEGA,
                                                            bi_graph, bi_surf, bi_w, EBI);
            final_scatter<<<gblk(EBI, BS), BS, 0, stream>>>(outp, H2, DEGB,
                                                            bi_surf, bi_graph, bi_w, EBI);
        }
    }
}